// GraphConvolution_3152505996094
// MI455X (gfx1250) — compile-verified
//
#include <hip/hip_runtime.h>
#include <hip/hip_bf16.h>

// GCN layer for MI455X (gfx1250, wave32):
//   support = x @ W            -- fp32 WMMA 16x16x4 GEMM, W transposed+padded in LDS
//   out     = segment_sum(support[col]*val, row) + b  -- wave-per-edge atomics
//
// fp32 WMMA keeps reference precision; the workload is L2/HBM bound
// (~620 MB of gather + atomic traffic vs 0.92 GFLOP), so low-precision
// matrix paths buy nothing here. v2: transposed LDS staging (B fragments are
// single ds_load_b64, no repack movs) + software-pipelined A loads.

typedef __attribute__((ext_vector_type(2))) float v2f;
typedef __attribute__((ext_vector_type(8))) float v8f;

#define DIM 96
#define KTILES 24        // 96 / 4
#define NTILES 6         // 96 / 16
#define LDSW 98          // padded transposed-row stride (dwords): 98%64=34,
                         // l16*34 mod 64 distinct for l16=0..15; even -> 8B-aligned b64
#define ROWS_PER_BLOCK 128  // 8 waves * 16 rows

__global__ __launch_bounds__(256)
void gcn_gemm_wmma(const float* __restrict__ x,
                   const float* __restrict__ w,
                   float* __restrict__ support,
                   int n_rows) {
    __shared__ float wlds[DIM * LDSW];  // 36.75 KB of 320 KB/WGP, transposed W

    const int tid = threadIdx.x;
    // Stage W transposed: wlds[c*LDSW + k] = W[k][c]. Global reads coalesced.
    #pragma unroll
    for (int i = tid; i < DIM * DIM; i += 256) {
        const int k = i / DIM;
        const int c = i - k * DIM;
        wlds[c * LDSW + k] = w[i];
    }
    __syncthreads();

    const int wave = tid >> 5;          // 0..7
    const int lane = tid & 31;
    const int half = lane >> 4;         // 0: K={k0,k0+1}, 1: K={k0+2,k0+3}
    const int l16  = lane & 15;

    const long row_base = (long)blockIdx.x * ROWS_PER_BLOCK + (long)wave * 16;

    // A-fragment row for this lane (ISA 16x4 f32 layout: M = lane%16 in both halves)
    long arow = row_base + l16;
    if (arow >= n_rows) arow = n_rows - 1;          // clamp; stores are guarded
    const float* __restrict__ xrow = x + arow * DIM;

    // Per-lane B base: column = t*16 + l16, K pair starts at half*2.
    const float* __restrict__ bbase = wlds + (l16 * LDSW + half * 2);

    v8f acc[NTILES];
    #pragma unroll
    for (int t = 0; t < NTILES; ++t) acc[t] = (v8f)0.0f;

    // Software-pipelined K loop: issue A(kt+1) before the WMMAs of kt.
    v2f a_cur;
    a_cur.x = xrow[half * 2];
    a_cur.y = xrow[half * 2 + 1];

    for (int kt = 0; kt < KTILES; ++kt) {
        v2f a_next = a_cur;
        if (kt < KTILES - 1) {
            const int ka = (kt + 1) * 4 + half * 2;
            a_next.x = xrow[ka];
            a_next.y = xrow[ka + 1];
        }
        const float* __restrict__ bk = bbase + kt * 4;
        #pragma unroll
        for (int t = 0; t < NTILES; ++t) {
            const v2f bfrag = *(const v2f*)(bk + t * 16 * LDSW);  // ds_load_b64
            acc[t] = __builtin_amdgcn_wmma_f32_16x16x4_f32(
                false, a_cur, false, bfrag, (short)0, acc[t], false, false);
        }
        a_cur = a_next;
    }

    // C/D 16x16 f32 layout: VGPR i -> M = i + half*8, N = lane%16.
    #pragma unroll
    for (int i = 0; i < 8; ++i) {
        const long r = row_base + i + half * 8;
        if (r < n_rows) {
            float* __restrict__ drow = support + r * DIM;
            #pragma unroll
            for (int t = 0; t < NTILES; ++t)
                drow[t * 16 + l16] = acc[t][i];
        }
    }
}

__global__ __launch_bounds__(256)
void gcn_init_out(float* __restrict__ out,
                  const float* __restrict__ bias,
                  long total) {
    long idx = (long)blockIdx.x * blockDim.x + threadIdx.x;
    if (idx < total) out[idx] = bias[idx % DIM];
}

__global__ __launch_bounds__(256)
void gcn_spmm_atomic(const int* __restrict__ erow,
                     const int* __restrict__ ecol,
                     const float* __restrict__ eval,
                     const float* __restrict__ support,
                     float* __restrict__ out,
                     int n_edges) {
    const int gwave = (int)(((long)blockIdx.x * blockDim.x + threadIdx.x) >> 5);
    const int lane  = threadIdx.x & 31;
    if (gwave >= n_edges) return;

    const int   r = erow[gwave];       // broadcast-coalesced scalar loads
    const int   c = ecol[gwave];
    const float v = eval[gwave];

    const float* __restrict__ src = support + (long)c * DIM;
    float* __restrict__ dst = out + (long)r * DIM;

    #pragma unroll
    for (int j = lane; j < DIM; j += 32)
        atomicAdd(dst + j, v * src[j]);   // global_atomic_add_f32, L2-resident
}

extern "C" void kernel_launch(void* const* d_in, const int* in_sizes, int n_in,
                              void* d_out, int out_size, void* d_ws, size_t ws_size,
                              hipStream_t stream) {
    const float* x    = (const float*)d_in[0];
    const int*   erow = (const int*)  d_in[1];
    const int*   ecol = (const int*)  d_in[2];
    const float* eval = (const float*)d_in[3];
    const float* w    = (const float*)d_in[4];
    const float* bias = (const float*)d_in[5];
    float* out = (float*)d_out;

    const int n_rows  = in_sizes[0] / DIM;   // 50000
    const int n_edges = in_sizes[1];         // 800000
    float* support = (float*)d_ws;           // n_rows*96 floats = 19.2 MB scratch

    // 1) support = x @ W   (fp32 WMMA)
    const int gemm_blocks = (n_rows + ROWS_PER_BLOCK - 1) / ROWS_PER_BLOCK;
    gcn_gemm_wmma<<<gemm_blocks, 256, 0, stream>>>(x, w, support, n_rows);

    // 2) out = b (broadcast) as segment-sum base
    const long total = (long)n_rows * DIM;
    const int init_blocks = (int)((total + 255) / 256);
    gcn_init_out<<<init_blocks, 256, 0, stream>>>(out, bias, total);

    // 3) out[row] += val * support[col]   (wave per edge)
    const int spmm_blocks = (n_edges + 7) / 8;   // 8 waves (edges) per 256-thread block
    gcn_spmm_atomic<<<spmm_blocks, 256, 0, stream>>>(erow, ecol, eval, support, out, n_edges);
}